// ErrorSampler_53876069761653
// MI455X (gfx1250) — compile-verified
//
#include <hip/hip_runtime.h>
#include <math.h>

#define B_     128
#define V_     128000
#define PLEN   2048
#define OLEN   256
#define CHUNK  8000      // 16 chunks * 8000 = 128000
#define NCHUNK 16
#define NBINS  4096
#define CAP    1024
#define TPB    512       // 16 wave32s
#define SMALL  128
#define LBUFN  8192      // CHUNK rounded up so every thread issues exactly 4 async b128 loads

typedef float    f32x4 __attribute__((ext_vector_type(4)));
typedef unsigned u32x4 __attribute__((ext_vector_type(4)));

__device__ __forceinline__ unsigned orderKey(float f) {
  unsigned u = __float_as_uint(f);
  return (u & 0x80000000u) ? ~u : (u | 0x80000000u);  // monotonic increasing with f
}

// ---- CDNA5 async global->LDS copy (16B per lane), tracked by ASYNCcnt ----
__device__ __forceinline__ void asyncLoad16(unsigned ldsByteAddr, const void* gptr) {
  asm volatile("global_load_async_to_lds_b128 %0, %1, off"
               :: "v"(ldsByteAddr), "v"((unsigned long long)(size_t)gptr)
               : "memory");
}
__device__ __forceinline__ void waitAsync0() {
  asm volatile("s_wait_asynccnt 0x0" ::: "memory");
}
__device__ __forceinline__ void waitAsync4() {   // allow the 4 newest (next chunk) in flight
  asm volatile("s_wait_asynccnt 0x4" ::: "memory");
}

// every thread issues exactly 4 async loads -> per-wave ASYNCcnt bump is uniform (4)
__device__ __forceinline__ void issueChunk(const float* __restrict__ rowLog, int base,
                                           float* buf, int tid) {
#pragma unroll
  for (int k = 0; k < 4; ++k) {
    int i4 = tid * 4 + k * (TPB * 4);   // 0..8188, step 4
    int g  = base + i4;
    if (g > V_ - 4) g = V_ - 4;         // clamp tail (lands in LDS pad region)
    asyncLoad16((unsigned)(size_t)(&buf[i4]), (const void*)(rowLog + g));
  }
}

// ---------------- Kernel A: non-temporal constant fill of outputs ----------------
__global__ void fill_kernel(float* __restrict__ probs, float* __restrict__ logprobs,
                            long long n) {
  long long i = (long long)blockIdx.x * blockDim.x + threadIdx.x;
  long long stride = (long long)gridDim.x * blockDim.x;
  f32x4 z = {0.f, 0.f, 0.f, 0.f};
  float ni = -INFINITY;
  f32x4 m = {ni, ni, ni, ni};
  long long n4 = n >> 2;  // n divisible by 4
  for (long long t = i; t < n4; t += stride) {
    __builtin_nontemporal_store(z, ((f32x4*)probs) + t);
    __builtin_nontemporal_store(m, ((f32x4*)logprobs) + t);
  }
}

// ---------------- Kernel B: one block per row ----------------
__global__ __launch_bounds__(TPB)
void sampler_kernel(const float* __restrict__ logits,
                    const int*   __restrict__ ptoks,
                    const int*   __restrict__ otoks,
                    const float* __restrict__ temps,
                    const float* __restrict__ topps,
                    const int*   __restrict__ topks,
                    const float* __restrict__ minps,
                    const float* __restrict__ presp,
                    const float* __restrict__ freqp,
                    const float* __restrict__ repp,
                    const int*   __restrict__ pert,
                    const float* __restrict__ gumbel,
                    float* __restrict__ outP,
                    float* __restrict__ outL,
                    float* __restrict__ outT) {
  __shared__ int      tokP[PLEN];
  __shared__ int      tokO[OLEN];
  __shared__ unsigned cnt[CHUNK];      // [15:0]=output count, bit16=prompt flag
  __shared__ float    lbuf0[LBUFN];    // double-buffered async staging
  __shared__ float    lbuf1[LBUFN];
  __shared__ unsigned hist[NBINS];
  __shared__ float    cval[CAP];
  __shared__ int      cidx[CAP];
  __shared__ unsigned part[32];
  __shared__ int      s_bstar;
  __shared__ int      s_ncand;
  __shared__ float    sE[SMALL];
  __shared__ int      sKeep[SMALL];
  __shared__ int      sFidx[SMALL];
  __shared__ float    sFprob[SMALL];
  __shared__ float    sFlog[SMALL];
  __shared__ float    sGsc[SMALL];
  __shared__ int      s_fcnt;

  const int row = blockIdx.x;
  const int tid = threadIdx.x;
  const float rep  = repp[row];
  const float fre  = freqp[row];
  const float pre  = presp[row];
  const float invT = 1.0f / temps[row];
  const int   kTop = topks[row];
  const float* rowLog = logits + (size_t)row * V_;

  for (int j = tid; j < PLEN;  j += TPB) tokP[j] = ptoks[row * PLEN + j];
  for (int j = tid; j < OLEN;  j += TPB) tokO[j] = otoks[row * OLEN + j];
  for (int j = tid; j < NBINS; j += TPB) hist[j] = 0u;
  if (tid == 0) s_ncand = 0;
  __syncthreads();

  // pass 0: histogram of adjusted logits; pass 1: collect candidates >= threshold bin
  for (int pass = 0; pass < 2; ++pass) {
    const int bstar = (pass == 1) ? s_bstar : 0;
    issueChunk(rowLog, 0, lbuf0, tid);            // prologue: fetch chunk 0
    for (int c = 0; c < NCHUNK; ++c) {
      float* cur = (c & 1) ? lbuf1 : lbuf0;
      float* nxt = (c & 1) ? lbuf0 : lbuf1;
      if (c + 1 < NCHUNK) issueChunk(rowLog, (c + 1) * CHUNK, nxt, tid);
      // zero counts (ds_store_b128)
      {
        u32x4 z4 = {0u, 0u, 0u, 0u};
        for (int i = tid * 4; i < CHUNK; i += TPB * 4) *(u32x4*)&cnt[i] = z4;
      }
      __syncthreads();
      // count tokens falling in this vocab chunk (LDS atomics)
      const int base = c * CHUNK;
      for (int j = tid; j < OLEN; j += TPB) {
        int t = tokO[j] - base;
        if ((unsigned)t < (unsigned)CHUNK) atomicAdd(&cnt[t], 1u);
      }
      for (int j = tid; j < PLEN; j += TPB) {
        int t = tokP[j] - base;
        if ((unsigned)t < (unsigned)CHUNK) atomicOr(&cnt[t], 0x10000u);
      }
      // wait for chunk c's data; next chunk's 4 loads stay in flight (in-order completion)
      if (c + 1 < NCHUNK) waitAsync4(); else waitAsync0();
      __syncthreads();
      // apply penalties + temperature; histogram or collect (vectorized LDS reads)
#pragma unroll
      for (int k = 0; k < 4; ++k) {
        int i4 = tid * 4 + k * (TPB * 4);
        if (i4 < CHUNK) {
          f32x4 lv = *(const f32x4*)&cur[i4];
          u32x4 cv = *(const u32x4*)&cnt[i4];
#pragma unroll
          for (int j = 0; j < 4; ++j) {
            float l = lv[j];
            unsigned cc = cv[j];
            unsigned oc = cc & 0xFFFFu;
            if (cc != 0u) l = (l > 0.0f) ? (l / rep) : (l * rep);
            l -= fre * (float)oc;
            if (oc) l -= pre;
            l *= invT;
            unsigned bin = orderKey(l) >> 20;
            if (pass == 0) {
              atomicAdd(&hist[bin], 1u);
            } else if ((int)bin >= bstar) {
              int pos = atomicAdd(&s_ncand, 1);
              if (pos < CAP) { cval[pos] = l; cidx[pos] = base + i4 + j; }
            }
          }
        }
      }
      __syncthreads();
    }
    if (pass == 0) {
      // find threshold bin b*: smallest set of top bins with cum count >= kTop
      if (tid < 32) {
        unsigned s = 0;
        for (int j = 0; j < NBINS / 32; ++j) s += hist[NBINS - 1 - (tid * (NBINS / 32) + j)];
        part[tid] = s;
      }
      __syncthreads();
      if (tid == 0) {
        unsigned need = (unsigned)kTop, cum = 0;
        int bsel = 0; bool done = false;
        for (int t = 0; t < 32 && !done; ++t) {
          if (cum + part[t] >= need) {
            for (int j = 0; j < NBINS / 32; ++j) {
              int b = NBINS - 1 - (t * (NBINS / 32) + j);
              cum += hist[b];
              if (cum >= need) { bsel = b; done = true; break; }
            }
          } else cum += part[t];
        }
        s_bstar = bsel;
      }
      __syncthreads();
    }
  }
  __syncthreads();

  // ---- bitonic sort of candidates (desc by value, asc by index) ----
  int n = s_ncand; if (n > CAP) n = CAP;
  for (int i = tid; i < CAP; i += TPB)
    if (i >= n) { cval[i] = -INFINITY; cidx[i] = 0x7FFFFFFF; }
  __syncthreads();
  for (int size = 2; size <= CAP; size <<= 1) {
    for (int stride = size >> 1; stride > 0; stride >>= 1) {
      int lo = tid & (stride - 1);
      int i  = ((tid - lo) << 1) + lo;
      int j  = i + stride;
      float av = cval[i], bv = cval[j];
      int   ai = cidx[i], bi = cidx[j];
      bool aFirst = (av > bv) || (av == bv && ai < bi);
      bool up = ((i & size) == 0);
      if (up ? !aFirst : aFirst) {
        cval[i] = bv; cval[j] = av;
        cidx[i] = bi; cidx[j] = ai;
      }
      __syncthreads();
    }
  }

  // ---- tiny epilogue on top-K survivors (K <= 99) ----
  if (tid == 0) {
    int K = kTop < n ? kTop : n;
    if (K > SMALL) K = SMALL;
    float m = cval[0];
    float S = 0.0f;
    for (int i = 0; i < K; ++i) { float e = __expf(cval[i] - m); sE[i] = e; S += e; }
    // top-p: keep iff exclusive prefix of p <= top_p ; min-p: keep iff e >= min_p (e0 = 1)
    float lim = topps[row] * S;
    float run = 0.0f;
    float mp  = minps[row];
    for (int i = 0; i < K; ++i) {
      bool k1 = (run <= lim);
      run += sE[i];
      sKeep[i] = (k1 && (sE[i] >= mp)) ? 1 : 0;
    }
    // top-2 gap masking (survivor 0 always kept so far)
    float S3 = 0.0f; int second = -1;
    for (int i = 0; i < K; ++i)
      if (sKeep[i]) { S3 += sE[i]; if (i > 0 && second < 0) second = i; }
    float gap = (1.0f - (second >= 0 ? sE[second] : 0.0f)) / S3;
    bool mask = (pert[row] < 3) && (gap < 0.9f);
    if (mask) sKeep[0] = 0;           // if masked, 'second' exists (gap would be 1 otherwise)
    int firstf = mask ? second : 0;
    float mf = cval[firstf];
    float Z = 0.0f; int fc = 0;
    for (int i = 0; i < K; ++i)
      if (sKeep[i]) {
        float ex = __expf(cval[i] - mf);
        sFidx[fc] = cidx[i];
        sFprob[fc] = ex;              // temp: e
        sFlog[fc]  = cval[i] - mf;    // temp: l - mf
        Z += ex; ++fc;
      }
    float invZ = 1.0f / Z;
    float lZ = __logf(Z);
    for (int j2 = 0; j2 < fc; ++j2) { sFprob[j2] *= invZ; sFlog[j2] -= lZ; }
    s_fcnt = fc;
  }
  __syncthreads();

  const int fc = s_fcnt;
  // gather gumbel only at survivor positions (parallel)
  if (tid < fc) sGsc[tid] = sFlog[tid] + gumbel[(size_t)row * V_ + sFidx[tid]];
  __syncthreads();
  if (tid == 0) {
    float bv = -INFINITY; int bi = 0x7FFFFFFF;
    for (int j2 = 0; j2 < fc; ++j2) {
      float s = sGsc[j2]; int vi = sFidx[j2];
      if (s > bv || (s == bv && vi < bi)) { bv = s; bi = vi; }
    }
    outT[row] = (float)bi;
  }
  // scatter survivor probs/logprobs over the constant-filled outputs
  if (tid < fc) {
    size_t o = (size_t)row * V_ + sFidx[tid];
    outP[o] = sFprob[tid];
    outL[o] = sFlog[tid];
  }
}

extern "C" void kernel_launch(void* const* d_in, const int* in_sizes, int n_in,
                              void* d_out, int out_size, void* d_ws, size_t ws_size,
                              hipStream_t stream) {
  const float* logits = (const float*)d_in[0];
  const int*   ptoks  = (const int*)  d_in[1];
  const int*   otoks  = (const int*)  d_in[2];
  const float* temps  = (const float*)d_in[3];
  const float* topps  = (const float*)d_in[4];
  const int*   topks  = (const int*)  d_in[5];
  const float* minps  = (const float*)d_in[6];
  const float* presp  = (const float*)d_in[7];
  const float* freqp  = (const float*)d_in[8];
  const float* repp   = (const float*)d_in[9];
  const int*   pert   = (const int*)  d_in[10];
  const float* gumbel = (const float*)d_in[11];

  float* outP = (float*)d_out;
  float* outL = outP + (size_t)B_ * V_;
  float* outT = outL + (size_t)B_ * V_;

  fill_kernel<<<2048, 256, 0, stream>>>(outP, outL, (long long)B_ * V_);
  sampler_kernel<<<B_, TPB, 0, stream>>>(logits, ptoks, otoks, temps, topps, topks,
                                         minps, presp, freqp, repp, pert, gumbel,
                                         outP, outL, outT);
}